// StandardRasterizer_29360396436094
// MI455X (gfx1250) — compile-verified
//
#include <hip/hip_runtime.h>

#define HH 224
#define WW 224
#define NB 4
#define NF 9976
#define NVERT 5023
#define NTPAD 9984            // 624 groups of 16
#define NGROUP 624
#define NSPLIT 8
#define GPS (NGROUP / NSPLIT) // 78 groups per split
#define TILE_W 16
#define TILE_H 4
#define TXN (WW / TILE_W)     // 14
#define TYN (HH / TILE_H)     // 56
#define TILES (TXN * TYN)     // 784
#define BIGF 1000000.0f
#define EPSF 1e-8f

typedef float wfloat2 __attribute__((ext_vector_type(2)));
typedef float wfloat8 __attribute__((ext_vector_type(8)));

// Exact replication of the reference vertex transform (f32, no FMA contraction,
// literal op order so rounding matches XLA's elementwise lowering).
__device__ __forceinline__ void xform(const float* __restrict__ vp,
                                      float& X, float& Y, float& Z) {
#pragma clang fp contract(off)
  float x = -vp[0];
  float y = -vp[1];
  float z = vp[2];
  x = x * 224.0f / 2.0f + 112.0f;
  y = y * 224.0f / 2.0f + 112.0f;
  x = 223.0f - x;
  y = 223.0f - y;
  x = -1.0f + (2.0f * x + 1.0f) / 224.0f;
  y = -1.0f + (2.0f * y + 1.0f) / 224.0f;
  x = x * 224.0f / 2.0f + 112.0f;
  y = y * 224.0f / 2.0f + 112.0f;
  z = z * 224.0f / 2.0f;
  X = x; Y = y; Z = z;
}

// ---------------- Pass 0: z-buffer init ----------------
__global__ void zinit_kernel(unsigned long long* __restrict__ zbuf) {
  int t = blockIdx.x * blockDim.x + threadIdx.x;
  if (t < NB * HH * WW) zbuf[t] = ~0ull;
}

// ---------------- Pass 1: per-triangle plane setup ----------------
// P0/P1: inv_den-scaled edge planes for w0,w1; P2: inv_depth plane.
__global__ void setup_kernel(const float* __restrict__ verts,
                             const int* __restrict__ faces,
                             float4* __restrict__ P0,
                             float4* __restrict__ P1,
                             float4* __restrict__ P2) {
#pragma clang fp contract(off)
  int t = blockIdx.x * blockDim.x + threadIdx.x;
  if (t >= NB * NTPAD) return;
  int b = t / NTPAD;
  int i = t - b * NTPAD;
  float4 p0 = make_float4(0.0f, 0.0f, -1.0f, 0.0f);  // force-miss
  float4 p1 = make_float4(0.0f, 0.0f, -1.0f, 0.0f);
  float4 p2 = make_float4(0.0f, 0.0f,  1.0f, 0.0f);
  if (i < NF) {
    const int* f = faces + (size_t)(b * NF + i) * 3;
    float x0, y0, z0, x1, y1, z1, x2, y2, z2;
    xform(verts + (size_t)(b * NVERT + f[0]) * 3, x0, y0, z0);
    xform(verts + (size_t)(b * NVERT + f[1]) * 3, x1, y1, z1);
    xform(verts + (size_t)(b * NVERT + f[2]) * 3, x2, y2, z2);
    float a0 = y1 - y2, b0 = x2 - x1;
    float a1 = y2 - y0, b1 = x0 - x2;
    float denom = a0 * (x0 - x2) + b0 * (y0 - y2);
    if (fabsf(denom) > EPSF) {
      float invden = 1.0f / denom;
      float iz0 = 1.0f / ((fabsf(z0) > EPSF) ? z0 : 1.0f);
      float iz1 = 1.0f / ((fabsf(z1) > EPSF) ? z1 : 1.0f);
      float iz2 = 1.0f / ((fabsf(z2) > EPSF) ? z2 : 1.0f);
      float c0 = -(a0 * x2 + b0 * y2);
      float c1 = -(a1 * x2 + b1 * y2);
      p0 = make_float4(a0 * invden, b0 * invden, c0 * invden, 0.0f);
      p1 = make_float4(a1 * invden, b1 * invden, c1 * invden, 0.0f);
      float d0 = iz0 - iz2, d1 = iz1 - iz2;
      p2 = make_float4(p0.x * d0 + p1.x * d1,
                       p0.y * d0 + p1.y * d1,
                       iz2 + p0.z * d0 + p1.z * d1, 0.0f);
    }
  }
  P0[t] = p0; P1[t] = p1; P2[t] = p2;
}

// ---------------- Pass 2: WMMA raster ----------------
// One wave per (batch, 16x4 tile, triangle-split). Per 16-triangle group and
// pixel row: 3x V_WMMA_F32_16X16X4_F32 evaluate w0, w1, inv_depth for
// 16 triangles x 16 pixels. Survivors fold into a per-lane (depthKey|id) u64
// min, merged cross-half by shuffle, committed with global atomicMin u64.
__global__ void __launch_bounds__(32)
raster_kernel(const float4* __restrict__ P0,
              const float4* __restrict__ P1,
              const float4* __restrict__ P2,
              unsigned long long* __restrict__ zbuf) {
  int blk   = blockIdx.x;
  int split = blk % NSPLIT;
  int rest  = blk / NSPLIT;
  int b     = rest / TILES;
  int tl    = rest - b * TILES;
  int ty    = tl / TXN;
  int tx    = tl - ty * TXN;
  int x0 = tx * TILE_W;
  int y0 = ty * TILE_H;

  int lane = threadIdx.x;
  int m    = lane & 15;
  int half = lane >> 4;

  // B matrix (4x16, K x N): row0 = px, row1 = py, row2 = 1, row3 = 0.
  // Lanes 0-15 hold K=0/1 in vgpr0/1; lanes 16-31 hold K=2/3.
  float Bx = half ? 1.0f : (float)(x0 + m);

  unsigned long long best[TILE_H];
#pragma unroll
  for (int r = 0; r < TILE_H; ++r) best[r] = ~0ull;

  int tribase = b * NTPAD + split * GPS * 16;
  int idbase0 = split * GPS * 16;

  for (int g = 0; g < GPS; ++g) {
    float4 q0 = P0[tribase + g * 16 + m];
    float4 q1 = P1[tribase + g * 16 + m];
    float4 q2 = P2[tribase + g * 16 + m];
    // A matrix (16x4, M x K): lanes 0-15 -> (a,b), lanes 16-31 -> (c,0).
    wfloat2 A0; A0.x = half ? q0.z : q0.x; A0.y = half ? 0.0f : q0.y;
    wfloat2 A1; A1.x = half ? q1.z : q1.x; A1.y = half ? 0.0f : q1.y;
    wfloat2 A2; A2.x = half ? q2.z : q2.x; A2.y = half ? 0.0f : q2.y;

#pragma unroll
    for (int row = 0; row < TILE_H; ++row) {
      float By = half ? 0.0f : (float)(y0 + row);
      wfloat2 Bm; Bm.x = Bx; Bm.y = By;
      wfloat8 cz = {};
      wfloat8 dw0 = __builtin_amdgcn_wmma_f32_16x16x4_f32(
          false, A0, false, Bm, (short)0, cz, false, false);
      wfloat8 dw1 = __builtin_amdgcn_wmma_f32_16x16x4_f32(
          false, A1, false, Bm, (short)0, cz, false, false);
      wfloat8 dzi = __builtin_amdgcn_wmma_f32_16x16x4_f32(
          false, A2, false, Bm, (short)0, cz, false, false);
      int idb = idbase0 + g * 16 + (half << 3);
#pragma unroll
      for (int r = 0; r < 8; ++r) {
        float w0 = dw0[r];
        float w1 = dw1[r];
        float w2 = 1.0f - w0 - w1;
        if (fminf(w0, fminf(w1, w2)) >= 0.0f) {   // s_cbranch_execz skip
          float invd = dzi[r];
          float gd = (fabsf(invd) > EPSF) ? invd : 1.0f;
          float depth = 1.0f / gd;
          if (depth < BIGF) {
            unsigned u = __float_as_uint(depth);
            unsigned key = u ^ (((unsigned)(((int)u) >> 31)) | 0x80000000u);
            unsigned long long pk =
                ((unsigned long long)key << 32) | (unsigned)(idb + r);
            if (pk < best[row]) best[row] = pk;
          }
        }
      }
    }
  }

#pragma unroll
  for (int row = 0; row < TILE_H; ++row) {
    unsigned lo = (unsigned)(best[row] & 0xFFFFFFFFull);
    unsigned hi = (unsigned)(best[row] >> 32);
    unsigned lo2 = __shfl_xor(lo, 16, 32);
    unsigned hi2 = __shfl_xor(hi, 16, 32);
    unsigned long long other = ((unsigned long long)hi2 << 32) | lo2;
    unsigned long long mn = (other < best[row]) ? other : best[row];
    if (half == 0 && mn != ~0ull) {
      atomicMin(&zbuf[((size_t)b * HH + (y0 + row)) * WW + (x0 + m)], mn);
    }
  }
}

// ---------------- Pass 3: resolve ----------------
// Recompute the winner's barycentrics with the reference's exact expression
// order (contract off), then the 3-term attribute dot in reference order.
__global__ void resolve_kernel(const float* __restrict__ verts,
                               const int* __restrict__ faces,
                               const float* __restrict__ attrs,
                               const unsigned long long* __restrict__ zbuf,
                               float* __restrict__ out) {
#pragma clang fp contract(off)
  int t = blockIdx.x * blockDim.x + threadIdx.x;
  if (t >= NB * HH * WW) return;
  int b = t / (HH * WW);
  int p = t - b * (HH * WW);
  int y = p / WW;
  int x = p - y * WW;

  float pv[16];
#pragma unroll
  for (int d = 0; d < 16; ++d) pv[d] = 0.0f;
  float vis = 0.0f;

  unsigned long long zv = zbuf[t];
  if (zv != ~0ull) {
    vis = 1.0f;
    int id = (int)(unsigned)(zv & 0xFFFFFFFFull);
    const int* f = faces + (size_t)(b * NF + id) * 3;
    float x0, y0, z0, x1, y1, z1, x2, y2, z2;
    xform(verts + (size_t)(b * NVERT + f[0]) * 3, x0, y0, z0);
    xform(verts + (size_t)(b * NVERT + f[1]) * 3, x1, y1, z1);
    xform(verts + (size_t)(b * NVERT + f[2]) * 3, x2, y2, z2);
    float a0 = y1 - y2, b0 = x2 - x1;
    float a1 = y2 - y0, b1 = x0 - x2;
    float denom = a0 * (x0 - x2) + b0 * (y0 - y2);
    bool ok = fabsf(denom) > EPSF;
    float invden = 1.0f / (ok ? denom : 1.0f);
    float pxf = (float)x, pyf = (float)y;
    float w0 = (a0 * (pxf - x2) + b0 * (pyf - y2)) * invden;
    float w1 = (a1 * (pxf - x2) + b1 * (pyf - y2)) * invden;
    float w2 = 1.0f - w0 - w1;
    float iz0 = 1.0f / ((fabsf(z0) > EPSF) ? z0 : 1.0f);
    float iz1 = 1.0f / ((fabsf(z1) > EPSF) ? z1 : 1.0f);
    float iz2 = 1.0f / ((fabsf(z2) > EPSF) ? z2 : 1.0f);
    float invd = w0 * iz0 + w1 * iz1 + w2 * iz2;
    float depth = 1.0f / ((fabsf(invd) > EPSF) ? invd : 1.0f);
    float bb0 = (w0 * iz0) * depth;
    float bb1 = (w1 * iz1) * depth;
    float bb2 = (w2 * iz2) * depth;
    const float* A = attrs + (size_t)((b * NF + id) * 3) * 16;
#pragma unroll
    for (int d = 0; d < 16; ++d) {
      pv[d] = bb0 * A[d] + bb1 * A[16 + d] + bb2 * A[32 + d];
    }
  }
#pragma unroll
  for (int d = 0; d < 16; ++d) {
    out[(((size_t)b * 17 + d) * HH + y) * WW + x] = pv[d];
  }
  out[(((size_t)b * 17 + 16) * HH + y) * WW + x] = vis;
}

extern "C" void kernel_launch(void* const* d_in, const int* in_sizes, int n_in,
                              void* d_out, int out_size, void* d_ws, size_t ws_size,
                              hipStream_t stream) {
  const float* verts = (const float*)d_in[0];
  const int*   faces = (const int*)d_in[1];
  const float* attrs = (const float*)d_in[2];
  float* out = (float*)d_out;

  char* ws = (char*)d_ws;
  unsigned long long* zbuf = (unsigned long long*)ws;
  size_t zbytes = (size_t)NB * HH * WW * sizeof(unsigned long long);
  float4* P0 = (float4*)(ws + zbytes);
  float4* P1 = P0 + (size_t)NB * NTPAD;
  float4* P2 = P1 + (size_t)NB * NTPAD;

  zinit_kernel<<<(NB * HH * WW) / 256, 256, 0, stream>>>(zbuf);
  setup_kernel<<<(NB * NTPAD) / 256, 256, 0, stream>>>(verts, faces, P0, P1, P2);
  raster_kernel<<<NB * TILES * NSPLIT, 32, 0, stream>>>(P0, P1, P2, zbuf);
  resolve_kernel<<<(NB * HH * WW) / 256, 256, 0, stream>>>(verts, faces, attrs,
                                                           zbuf, out);
}